// SEIDR_44452911513662
// MI455X (gfx1250) — compile-verified
//
#include <hip/hip_runtime.h>

// SEIDR linear recurrence, parallelized as an associative scan over 8x8 f32
// transition matrices. Matrix composition uses CDNA5 V_WMMA_F32_16X16X4_F32
// (full-fp32 WMMA -> no precision loss vs scalar f32 FMA reassociation).

typedef __attribute__((ext_vector_type(2))) float v2f;
typedef __attribute__((ext_vector_type(8))) float v8f;

constexpr int NCHUNK = 256;  // chunks in the parallel scan
constexpr int NGROUP = 32;   // groups of 8 chunks

struct Params {
  const float *D0, *Di0, *Di1, *R0, *Ri0, *Ri1, *A0, *Ai0, *Ai1;
  const float *af, *gm, *ep, *di, *k, *lf;
};

// ---------------------------------------------------------------------------
// Phase 1: per-chunk product of step matrices. One lane per matrix COLUMN
// (linearity => applying the sparse row-update formulas to a basis column is
// exact). 256 chunks * 8 columns = 2048 lanes; ~12 VALU ops per step.
// Row/state order: 0:D 1:Di 2:R 3:Ri 4:A 5:Ai 6:I 7:E. wsP[c][row*8+col].
// ---------------------------------------------------------------------------
__global__ void seidr_chunk_mats(Params P, float* __restrict__ wsP, int S, int Lm1) {
  const int gid = blockIdx.x * blockDim.x + threadIdx.x;
  const int chunk = gid >> 3;
  const int col = gid & 7;
  if (chunk >= NCHUNK) return;
  const float af = *P.af, gm = *P.gm, ep = *P.ep, di = *P.di, k = *P.k;
  const float gmaf = gm + af;
  float D  = (col == 0) ? 1.f : 0.f;
  float Di = (col == 1) ? 1.f : 0.f;
  float R  = (col == 2) ? 1.f : 0.f;
  float Ri = (col == 3) ? 1.f : 0.f;
  float A  = (col == 4) ? 1.f : 0.f;
  float Ai = (col == 5) ? 1.f : 0.f;
  float I  = (col == 6) ? 1.f : 0.f;
  float E  = (col == 7) ? 1.f : 0.f;
  const int t0 = chunk * S;
  const int tEnd = min(t0 + S, Lm1);
  for (int t = t0; t < tEnd; ++t) {
    const float f  = P.lf[t];
    const float D1 = D + Di;
    const float R1 = R + Ri;
    const float A1 = A + Ai;
    const float In = I + (ep * E - di * I);
    const float Ri1 = gm * D1;
    const float Ai1 = af * D1;
    const float Di1 = di * In - gmaf * D1;
    const float E1  = E + f * (k * E + I) - ep * E;  // uses OLD I
    D = D1; Di = Di1; R = R1; Ri = Ri1; A = A1; Ai = Ai1; I = In; E = E1;
  }
  float* m = wsP + (size_t)chunk * 64;
  m[0 * 8 + col] = D;  m[1 * 8 + col] = Di;
  m[2 * 8 + col] = R;  m[3 * 8 + col] = Ri;
  m[4 * 8 + col] = A;  m[5 * 8 + col] = Ai;
  m[6 * 8 + col] = I;  m[7 * 8 + col] = E;
}

// ---------------------------------------------------------------------------
// Phase 2 (single block, 256 threads = 8 wave32):
//  B) each wave sequentially composes TWO independent groups of 8 chunk
//     matrices at once: pack the two 8x8 products into one 16x16x8 WMMA
//     computation (A = [L0;L1] 16x8, B = [R0 R1] 8x16; diagonal blocks of C
//     are the products). K=8 -> two chained v_wmma_f32_16x16x4_f32.
//     Running products live in LDS (same-wave LDS ordering => no barriers).
//  C) one thread scans the 32 group products with sequential 8x8 matvecs
//     to get group-boundary states.
//  D) 32 threads expand to per-chunk boundary states in wsS.
// ---------------------------------------------------------------------------
__global__ void seidr_scan(Params P, const float* __restrict__ wsP,
                           float* __restrict__ wsS) {
  __shared__ float Gs[NGROUP * 64];        // running group products, row-major
  __shared__ float Ts[(NGROUP + 1) * 8];   // group-boundary states
  const int tid = threadIdx.x;

  // A) seed each group's running product with its first chunk matrix
  for (int i = tid; i < NGROUP * 64; i += blockDim.x) {
    const int g = i >> 6, off = i & 63;
    Gs[i] = wsP[(size_t)(g * 8) * 64 + off];
  }
  __syncthreads();

  // B) WMMA composition chains (wave-private groups)
  const int l = tid & 31, w = tid >> 5;
  const int h = l >> 4;    // K-half select (lanes 16-31 hold K+2)
  const int q = l & 15;    // M index for A, N index for B
  for (int tsk = 0; tsk < 2; ++tsk) {
    const int wt = w * 2 + tsk;          // 16 wave-tasks
    const int g0 = wt * 2, g1 = g0 + 1;  // two independent groups per task
    for (int j = 1; j < 8; ++j) {
      // A operand: next matrix L (stacked [L0;L1], row = q), from global wsP
      const float* Lrow = (q < 8)
          ? (wsP + (size_t)(g0 * 8 + j) * 64 + q * 8)
          : (wsP + (size_t)(g1 * 8 + j) * 64 + (q - 8) * 8);
      v2f a01 = { Lrow[0 + 2 * h], Lrow[1 + 2 * h] };   // K = 0,1 (+2h)
      v2f a23 = { Lrow[4 + 2 * h], Lrow[5 + 2 * h] };   // K = 4,5 (+2h)
      // B operand: running products [R0 R1] (col = q), from LDS
      const float* Bcol = (q < 8) ? (&Gs[g0 * 64 + q]) : (&Gs[g1 * 64 + (q - 8)]);
      v2f b01 = { Bcol[(0 + 2 * h) * 8], Bcol[(1 + 2 * h) * 8] };
      v2f b23 = { Bcol[(4 + 2 * h) * 8], Bcol[(5 + 2 * h) * 8] };
      v8f c = {0.f, 0.f, 0.f, 0.f, 0.f, 0.f, 0.f, 0.f};
      c = __builtin_amdgcn_wmma_f32_16x16x4_f32(false, a01, false, b01,
                                                (short)0, c, false, false);
      c = __builtin_amdgcn_wmma_f32_16x16x4_f32(false, a23, false, b23,
                                                (short)0, c, false, false);
      // Write back diagonal blocks: C vgpr r = row (M=r+8h), lane = col N.
      if (l < 8) {            // product 0: M<8 (h=0), N=l<8
        #pragma unroll
        for (int r = 0; r < 8; ++r) Gs[g0 * 64 + r * 8 + l] = c[r];
      } else if (l >= 24) {   // product 1: M>=8 (h=1), N=l-16>=8
        #pragma unroll
        for (int r = 0; r < 8; ++r) Gs[g1 * 64 + r * 8 + (l - 24)] = c[r];
      }
    }
  }
  __syncthreads();

  // C) sequential group-boundary state scan (32 tiny matvecs)
  if (tid == 0) {
    const float di = *P.di, ep = *P.ep;
    float s[8];
    s[0] = *P.D0; s[1] = *P.Di0; s[2] = *P.R0; s[3] = *P.Ri0;
    s[4] = *P.A0; s[5] = *P.Ai0;
    const float I0  = (*P.Di0 + *P.Ri0 + *P.Ai0) / di;
    const float I1t = (*P.Di1 + *P.Ri1 + *P.Ai1) / di;
    s[6] = I0;
    s[7] = (I1t - I0 + di * I0) / ep;
    #pragma unroll
    for (int r = 0; r < 8; ++r) Ts[r] = s[r];
    for (int g = 0; g < NGROUP; ++g) {
      float y[8];
      #pragma unroll
      for (int r = 0; r < 8; ++r) {
        float acc = 0.f;
        #pragma unroll
        for (int cc = 0; cc < 8; ++cc) acc += Gs[g * 64 + r * 8 + cc] * s[cc];
        y[r] = acc;
      }
      #pragma unroll
      for (int r = 0; r < 8; ++r) { s[r] = y[r]; Ts[(g + 1) * 8 + r] = y[r]; }
    }
  }
  __syncthreads();

  // D) per-chunk boundary states (32 threads, 8 matvecs each)
  if (tid < NGROUP) {
    const int g = tid;
    float s[8];
    #pragma unroll
    for (int r = 0; r < 8; ++r) s[r] = Ts[g * 8 + r];
    for (int j = 0; j < 8; ++j) {
      const int c = g * 8 + j;
      #pragma unroll
      for (int r = 0; r < 8; ++r) wsS[(size_t)c * 8 + r] = s[r];
      float y[8];
      #pragma unroll
      for (int r = 0; r < 8; ++r) {
        float acc = 0.f;
        const float* Mr = wsP + (size_t)c * 64 + r * 8;
        #pragma unroll
        for (int cc = 0; cc < 8; ++cc) acc += Mr[cc] * s[cc];
        y[r] = acc;
      }
      #pragma unroll
      for (int r = 0; r < 8; ++r) s[r] = y[r];
    }
  }
}

// ---------------------------------------------------------------------------
// Phase 3: replay the scalar recurrence per chunk from its boundary state,
// using the reference's exact expression order, emitting D/R/A for every t.
// ---------------------------------------------------------------------------
__global__ void seidr_emit(Params P, const float* __restrict__ wsS,
                           float* __restrict__ out, int S, int L) {
  const int c = blockIdx.x * blockDim.x + threadIdx.x;
  if (c >= NCHUNK) return;
  const float af = *P.af, gm = *P.gm, ep = *P.ep, di = *P.di, k = *P.k;
  const float gmaf = gm + af;
  const float* s = wsS + (size_t)c * 8;
  float D = s[0], Di = s[1], R = s[2], Ri = s[3];
  float A = s[4], Ai = s[5], I = s[6], E = s[7];
  const int t0 = c * S;
  for (int i = 0; i < S; ++i) {
    const int t = t0 + i;
    if (t >= L) break;
    out[t] = D; out[L + t] = R; out[2 * L + t] = A;
    if (t < L - 1) {
      const float f  = P.lf[t];
      const float D1 = D + Di;
      const float R1 = R + Ri;
      const float A1 = A + Ai;
      const float In = I + (ep * E - di * I);
      const float Ri1 = gm * D1;
      const float Ai1 = af * D1;
      const float Di1 = di * In - gmaf * D1;
      const float E1  = E + f * (k * E + I) - ep * E;
      D = D1; Di = Di1; R = R1; Ri = Ri1; A = A1; Ai = Ai1; I = In; E = E1;
    }
  }
}

extern "C" void kernel_launch(void* const* d_in, const int* in_sizes, int n_in,
                              void* d_out, int out_size, void* d_ws, size_t ws_size,
                              hipStream_t stream) {
  Params P;
  P.D0  = (const float*)d_in[0];  P.Di0 = (const float*)d_in[1];
  P.Di1 = (const float*)d_in[2];  P.R0  = (const float*)d_in[3];
  P.Ri0 = (const float*)d_in[4];  P.Ri1 = (const float*)d_in[5];
  P.A0  = (const float*)d_in[6];  P.Ai0 = (const float*)d_in[7];
  P.Ai1 = (const float*)d_in[8];  P.af  = (const float*)d_in[9];
  P.gm  = (const float*)d_in[10]; P.ep  = (const float*)d_in[11];
  P.di  = (const float*)d_in[12]; P.k   = (const float*)d_in[13];
  P.lf  = (const float*)d_in[14];

  const int Lm1 = in_sizes[14];              // 99999
  const int L   = Lm1 + 1;                   // 100000
  const int S   = (Lm1 + NCHUNK - 1) / NCHUNK;

  float* wsP = (float*)d_ws;                 // NCHUNK * 64 floats (64 KB)
  float* wsS = wsP + (size_t)NCHUNK * 64;    // NCHUNK * 8 floats (8 KB)

  seidr_chunk_mats<<<(NCHUNK * 8) / 256, 256, 0, stream>>>(P, wsP, S, Lm1);
  seidr_scan<<<1, 256, 0, stream>>>(P, wsP, wsS);
  seidr_emit<<<NCHUNK / 32, 32, 0, stream>>>(P, wsS, (float*)d_out, S, L);
}